// NoisyTopkRouter_15659450761985
// MI455X (gfx1250) — compile-verified
//
#include <hip/hip_runtime.h>
#include <math.h>

typedef __attribute__((ext_vector_type(2))) float v2f;
typedef __attribute__((ext_vector_type(4))) float v4f;
typedef __attribute__((ext_vector_type(8))) float v8f;

#define B_ 128
#define T_ 1024   // T1*T2
#define D_ 1024
#define E_ 64
#define TSPLIT 16

// ---------------------------------------------------------------- zero ws
__global__ void zero_ws_kernel(float* __restrict__ p, int n) {
    int i = blockIdx.x * blockDim.x + threadIdx.x;
    if (i < n) p[i] = 0.0f;
}

// ---------------------------------------------------------------- pooling
// mean over T (=T1*T2) axis: [B, T, D] -> [B, D].  This phase reads 512 MiB
// once (the whole problem is bound by it: ~22us at 23.3 TB/s).  The stream
// has zero reuse and exceeds the 192 MB L2, so loads are marked non-temporal
// to keep pooled/W tensors resident for the WMMA kernel that follows.
// Coalesced b128 loads; TSPLIT blocks per batch row; partial sums merged
// with f32 global atomics (8 MB of atomic traffic vs 512 MB of reads).
__global__ void pool_kernel(const float* __restrict__ mh,
                            float* __restrict__ pooled) {
    const int b   = blockIdx.x;
    const int ts  = blockIdx.y;
    const int tid = threadIdx.x;            // 256 threads, 4 floats each = D
    const int d   = tid * 4;
    const int rows = T_ / TSPLIT;           // 64 t-rows per block

    const v4f* base =
        (const v4f*)(mh + ((size_t)b * T_ + (size_t)ts * rows) * D_ + d);
    float sx = 0.f, sy = 0.f, sz = 0.f, sw = 0.f;
#pragma unroll 8
    for (int t = 0; t < rows; ++t) {
        v4f v = __builtin_nontemporal_load(base + (size_t)t * (D_ / 4));
        sx += v.x; sy += v.y; sz += v.z; sw += v.w;
    }
    const float inv = 1.0f / (float)T_;
    float* out = pooled + b * D_ + d;
    atomicAdd(out + 0, sx * inv);
    atomicAdd(out + 1, sy * inv);
    atomicAdd(out + 2, sz * inv);
    atomicAdd(out + 3, sw * inv);
}

// ---------------------------------------------------------------- WMMA GEMM
// route = pooled @ W_route^T, noise = pooled @ W_noise^T
// pooled: [128,1024]  W: [64,1024]  out: [128,64]
// One wave per 16x16 output tile; V_WMMA_F32_16X16X4_F32, K-loop of 256.
// A 16x4 layout: lane l%16 = M, float2 = K{0,1} (lanes 0-15) / K{2,3} (16-31).
// B 4x16 layout mirrors with lane = N; B[k][n] = W[n][k] so both are
// contiguous float2 loads along the K (fast) axis.
__global__ void gemm_wmma_kernel(const float* __restrict__ pooled,
                                 const float* __restrict__ W_route,
                                 const float* __restrict__ W_noise,
                                 float* __restrict__ out_route,
                                 float* __restrict__ out_noise) {
    const int tile = blockIdx.x;     // 0..31
    const int tm   = tile >> 2;      // 8 M-tiles over B=128
    const int tn   = tile & 3;       // 4 N-tiles over E=64
    const int lane = threadIdx.x;    // 0..31 (wave32)
    const int half = lane >> 4;      // K sub-pair selector
    const int l    = lane & 15;

    const float* Ab  = pooled  + (size_t)(tm * 16 + l) * D_ + half * 2;
    const float* Brb = W_route + (size_t)(tn * 16 + l) * D_ + half * 2;
    const float* Bnb = W_noise + (size_t)(tn * 16 + l) * D_ + half * 2;

    v8f accR = {};
    v8f accN = {};
#pragma unroll 4
    for (int k = 0; k < D_; k += 4) {
        v2f a  = *(const v2f*)(Ab  + k);
        v2f br = *(const v2f*)(Brb + k);
        v2f bn = *(const v2f*)(Bnb + k);
        accR = __builtin_amdgcn_wmma_f32_16x16x4_f32(
            false, a, false, br, (short)0, accR, false, false);
        accN = __builtin_amdgcn_wmma_f32_16x16x4_f32(
            false, a, false, bn, (short)0, accN, false, false);
    }

    // C/D layout: VGPR r -> M = r + 8*half, N = lane%16
#pragma unroll
    for (int r = 0; r < 8; ++r) {
        const int m = tm * 16 + r + half * 8;
        const int n = tn * 16 + l;
        out_route[m * E_ + n] = accR[r];
        out_noise[m * E_ + n] = accN[r];
    }
}

// ---------------------------------------------------------------- epilogue
__device__ __forceinline__ float blk_max64(float v, float* red, int tid) {
    __syncthreads();
    red[tid] = v;
    __syncthreads();
    for (int s = E_ / 2; s > 0; s >>= 1) {
        if (tid < s) red[tid] = fmaxf(red[tid], red[tid + s]);
        __syncthreads();
    }
    float r = red[0];
    __syncthreads();
    return r;
}

__device__ __forceinline__ float blk_sum64(float v, float* red, int tid) {
    __syncthreads();
    red[tid] = v;
    __syncthreads();
    for (int s = E_ / 2; s > 0; s >>= 1) {
        if (tid < s) red[tid] = red[tid] + red[tid + s];
        __syncthreads();
    }
    float r = red[0];
    __syncthreads();
    return r;
}

__global__ void epilogue_kernel(const float* __restrict__ gr,
                                const float* __restrict__ gn,
                                const float* __restrict__ b_route,
                                const float* __restrict__ b_noise,
                                const float* __restrict__ noise_eps,
                                const int* __restrict__ mis_mask,
                                float* __restrict__ out) {
    const int b = blockIdx.x;    // row
    const int e = threadIdx.x;   // 0..63

    __shared__ float red[E_];
    __shared__ float s_noisy[E_];

    const float lr = gr[b * E_ + e] + b_route[e];
    const float ln = gn[b * E_ + e] + b_noise[e];

    // logits = softmax(lr)
    float m  = blk_max64(lr, red, e);
    float el = expf(lr - m);
    float s  = blk_sum64(el, red, e);
    const float logits = el / s;

    // noise = softmax(noise_eps * softplus(ln))
    const float sp = fmaxf(ln, 0.0f) + log1pf(expf(-fabsf(ln)));
    const float z  = noise_eps[b * E_ + e] * sp;
    m = blk_max64(z, red, e);
    float ez = expf(z - m);
    s = blk_sum64(ez, red, e);
    const float noise = ez / s;

    const float noisy = logits + noise;
    s_noisy[e] = noisy;
    __syncthreads();

    // stable descending rank (matches jax.lax.top_k tie-break by index)
    int rank = 0;
#pragma unroll 8
    for (int j = 0; j < E_; ++j) {
        const float vj = s_noisy[j];
        rank += (vj > noisy) || (vj == noisy && j < e);
    }

    const int k = mis_mask[b];
    float sparse = (rank < k) ? noisy : 0.0f;
    // torch scatter pad quirk: expert 0 forced to 0 when k < E
    if (e == 0 && k < E_) sparse = 0.0f;

    // router_output = softmax(sparse)
    m = blk_max64(sparse, red, e);
    float es = expf(sparse - m);
    s = blk_sum64(es, red, e);

    out[b * E_ + e]            = es / s;   // router_output
    out[B_ * E_ + b * E_ + e]  = noisy;    // noisy (2nd tuple element)
}

// ---------------------------------------------------------------- launch
extern "C" void kernel_launch(void* const* d_in, const int* in_sizes, int n_in,
                              void* d_out, int out_size, void* d_ws, size_t ws_size,
                              hipStream_t stream) {
    const float* mh        = (const float*)d_in[0];
    const float* W_route   = (const float*)d_in[1];
    const float* b_route   = (const float*)d_in[2];
    const float* W_noise   = (const float*)d_in[3];
    const float* b_noise   = (const float*)d_in[4];
    const float* noise_eps = (const float*)d_in[5];
    const int*   mis_mask  = (const int*)d_in[6];
    float* out = (float*)d_out;

    float* pooled = (float*)d_ws;           // [128*1024]
    float* grt    = pooled + B_ * D_;       // [128*64]
    float* gns    = grt + B_ * E_;          // [128*64]

    zero_ws_kernel<<<(B_ * D_ + 255) / 256, 256, 0, stream>>>(pooled, B_ * D_);
    pool_kernel<<<dim3(B_, TSPLIT), 256, 0, stream>>>(mh, pooled);
    gemm_wmma_kernel<<<32, 32, 0, stream>>>(pooled, W_route, W_noise, grt, gns);
    epilogue_kernel<<<B_, E_, 0, stream>>>(grt, gns, b_route, b_noise,
                                           noise_eps, mis_mask, out);
}